// TrainableInvertibleLinear_68118181315015
// MI455X (gfx1250) — compile-verified
//
#include <hip/hip_runtime.h>

#define DIM   128
#define PAD   132          // padded LDS row stride in floats (128 + 4) -> bank-conflict free
#define BATCH 1048576
#define EPSF  1e-8f

typedef __attribute__((ext_vector_type(2))) float v2f;
typedef __attribute__((ext_vector_type(8))) float v8f;
typedef int v4i __attribute__((__vector_size__(4 * sizeof(int))));

// ---- gfx1250 async global->LDS path (probe via __has_builtin, safe fallback) ----
#if defined(__has_builtin)
# if __has_builtin(__builtin_amdgcn_global_load_async_to_lds_b128)
#  define HAVE_ASYNC_B128 1
# endif
# if __has_builtin(__builtin_amdgcn_s_wait_asynccnt)
#  define HAVE_WAIT_ASYNC 1
# endif
#endif

// Exact parameter types per hipcc diagnostic:
//   arg0: v4i in address space 1 (global), non-const
//   arg1: v4i in address space 3 (LDS)
#define GLB_V4I(p) ((__attribute__((address_space(1))) v4i*)(p))
#define LDS_V4I(p) ((__attribute__((address_space(3))) v4i*)(p))

__device__ __forceinline__ void wait_async_zero()
{
#if defined(HAVE_WAIT_ASYNC)
    __builtin_amdgcn_s_wait_asynccnt(0);
#else
    asm volatile("s_wait_asynccnt 0" ::: "memory");
#endif
}

// ---------------------------------------------------------------------------
// Kernel A: build W^T = (Q * diag(diag_scale*scale))^T and logdet.
// Q is held entirely in registers: thread t=(r,c8) owns Q[r][c8*16 .. c8*16+15].
// The 8 threads of a row are consecutive lanes of one wave -> shuffle reduction,
// no barriers inside the 128-step Householder loop.
// ---------------------------------------------------------------------------
__global__ __launch_bounds__(1024)
void build_wt_kernel(const float* __restrict__ v_vectors,
                     const float* __restrict__ scale,
                     const float* __restrict__ diag_scale,
                     float*       __restrict__ wt,      // [128][128] = W^T
                     float*       __restrict__ logdet)
{
    __shared__ float V[DIM * PAD];   // V[i][k] = (normalized) column i of v_vectors

    const int t  = threadIdx.x;      // 0..1023
    const int r  = t >> 3;           // row of Q owned: 0..127
    const int c8 = t & 7;            // 16-column chunk: 0..7
    const int k0 = c8 * 16;

    // Load v_vectors transposed into LDS (coalesced global reads):
    // V[i][k] = v_vectors[k][i]  (column i of v_vectors = scan vector i)
    for (int g = t; g < DIM * DIM; g += 1024) {
        const int k = g >> 7;        // global row
        const int i = g & 127;       // global col
        V[i * PAD + k] = v_vectors[g];
    }
    __syncthreads();

    // Normalize row r of V: partial sum of squares over 16 elems, 8-lane reduce.
    {
        float ps = 0.f;
        #pragma unroll
        for (int j = 0; j < 16; ++j) {
            const float v = V[r * PAD + k0 + j];
            ps += v * v;
        }
        ps += __shfl_xor(ps, 1);
        ps += __shfl_xor(ps, 2);
        ps += __shfl_xor(ps, 4);
        const float inv = 1.0f / (sqrtf(ps) + EPSF);
        #pragma unroll
        for (int j = 0; j < 16; ++j) V[r * PAD + k0 + j] *= inv;
    }
    __syncthreads();

    // Q = I, in registers.
    float q[16];
    #pragma unroll
    for (int j = 0; j < 16; ++j) q[j] = (r == k0 + j) ? 1.0f : 0.0f;

    // 128 Householder steps: Q <- Q - 2 (Q v_i) v_i^T.  No barriers needed:
    // each thread touches only its own row chunk; reduction is intra-wave.
    for (int i = 0; i < DIM; ++i) {
        const float* __restrict__ vrow = &V[i * PAD + k0];
        float p = 0.f;
        #pragma unroll
        for (int j = 0; j < 16; ++j) p += q[j] * vrow[j];
        p += __shfl_xor(p, 1);
        p += __shfl_xor(p, 2);
        p += __shfl_xor(p, 4);
        const float t2 = 2.0f * p;
        #pragma unroll
        for (int j = 0; j < 16; ++j) q[j] -= t2 * vrow[j];
    }

    // W[r][c] = Q[r][c] * diag_scale[c] * s  ->  store transposed: wt[c][r]
    const float s0 = scale[0];
    #pragma unroll
    for (int j = 0; j < 16; ++j) {
        const int c = k0 + j;
        wt[c * DIM + r] = q[j] * diag_scale[c] * s0;
    }

    if (t == 0) {
        float acc = 0.f;
        for (int c = 0; c < DIM; ++c)
            acc += logf(fabsf(diag_scale[c] * s0) + EPSF);
        *logdet = acc;
    }
}

// ---------------------------------------------------------------------------
// Kernel B: y = x @ W via V_WMMA_F32_16X16X4_F32.
// Block = 256 threads (8 waves); each block computes a 128x128 output tile.
// LDS fill uses GLOBAL_LOAD_ASYNC_TO_LDS_B128 (ASYNCcnt) when the toolchain
// exposes it; otherwise the VGPR-staged b128+ds_store path.
// Output is transposed through the (wave-private) Xs strip and written as
// contiguous 512 B global_store_b128 per instruction.
// ---------------------------------------------------------------------------
__global__ __launch_bounds__(256)
void gemm_wmma_kernel(const float* __restrict__ x,
                      const float* __restrict__ wt,   // W^T [n][k]
                      float*       __restrict__ y)
{
    __shared__ float Ws[DIM * PAD];  // Ws[n][k] = W^T, padded stride
    __shared__ float Xs[DIM * PAD];  // Xs[m][k], padded stride

    const int    tid     = threadIdx.x;
    const size_t rowBase = (size_t)blockIdx.x * DIM;

    const float4* __restrict__ wt4 = (const float4*)wt;
    const float4* __restrict__ x4  = (const float4*)(x + rowBase * DIM);

#if defined(HAVE_ASYNC_B128)
    #pragma unroll
    for (int it = 0; it < 16; ++it) {
        const int q   = tid + it * 256;   // quad index 0..4095
        const int row = q >> 5;
        const int kq  = q & 31;
        __builtin_amdgcn_global_load_async_to_lds_b128(
            GLB_V4I(wt4 + q), LDS_V4I(&Ws[row * PAD + kq * 4]), 0, 0);
        __builtin_amdgcn_global_load_async_to_lds_b128(
            GLB_V4I(x4 + q), LDS_V4I(&Xs[row * PAD + kq * 4]), 0, 0);
    }
    wait_async_zero();
#else
    #pragma unroll
    for (int it = 0; it < 16; ++it) {
        const int q   = tid + it * 256;
        const int row = q >> 5;
        const int kq  = q & 31;
        const float4 wv = wt4[q];
        const float4 xv = x4[q];
        *(float4*)&Ws[row * PAD + kq * 4] = wv;
        *(float4*)&Xs[row * PAD + kq * 4] = xv;
    }
#endif
    __syncthreads();

    const int wave = tid >> 5;
    const int lane = tid & 31;
    const int lh   = lane & 15;
    const int hi   = lane >> 4;

    const float* __restrict__ arow = &Xs[(wave * 16 + lh) * PAD + 2 * hi];

    const v8f zero = {0.f, 0.f, 0.f, 0.f, 0.f, 0.f, 0.f, 0.f};
    v8f acc[8];
    #pragma unroll
    for (int j = 0; j < 8; ++j) acc[j] = zero;

    #pragma unroll 4
    for (int k0 = 0; k0 < DIM; k0 += 4) {
        const v2f a = *(const v2f*)(arow + k0);
        #pragma unroll
        for (int j = 0; j < 8; ++j) {
            const v2f b = *(const v2f*)&Ws[(j * 16 + lh) * PAD + k0 + 2 * hi];
            acc[j] = __builtin_amdgcn_wmma_f32_16x16x4_f32(
                /*neg_a=*/false, a, /*neg_b=*/false, b,
                /*c_mod=*/(short)0, acc[j],
                /*reuse_a=*/false, /*reuse_b=*/false);
        }
    }

    // Transpose D fragments through the wave-private Xs strip (no barrier:
    // only this wave ever touches rows [wave*16, wave*16+16) and the compiler
    // orders the ds traffic with DScnt).  D layout: VGPR v holds M=v (lanes
    // 0-15) and M=v+8 (lanes 16-31), N=lh.
    #pragma unroll
    for (int j = 0; j < 8; ++j) {
        #pragma unroll
        for (int v = 0; v < 8; ++v) {
            Xs[(wave * 16 + hi * 8 + v) * PAD + j * 16 + lh] = acc[j][v];
        }
    }

    // Coalesced writeback: each instruction stores one full 512 B row of y.
    const float* __restrict__ src  = &Xs[(wave * 16) * PAD];
    float*       __restrict__ yrow = y + (rowBase + (size_t)(wave * 16)) * DIM;
    #pragma unroll
    for (int it = 0; it < 16; ++it) {
        const float4 val = *(const float4*)&src[it * PAD + lane * 4];
        *(float4*)&yrow[(size_t)it * DIM + lane * 4] = val;
    }
}

// ---------------------------------------------------------------------------
extern "C" void kernel_launch(void* const* d_in, const int* in_sizes, int n_in,
                              void* d_out, int out_size, void* d_ws, size_t ws_size,
                              hipStream_t stream)
{
    (void)in_sizes; (void)n_in; (void)out_size; (void)ws_size;

    const float* x          = (const float*)d_in[0];
    const float* v_vectors  = (const float*)d_in[1];
    const float* scale      = (const float*)d_in[2];
    const float* diag_scale = (const float*)d_in[3];

    float* y      = (float*)d_out;
    float* logdet = y + (size_t)BATCH * DIM;   // tuple output: y flat, then logdet
    float* wt     = (float*)d_ws;              // 64 KB scratch for W^T

    build_wt_kernel<<<1, 1024, 0, stream>>>(v_vectors, scale, diag_scale, wt, logdet);
    gemm_wmma_kernel<<<BATCH / DIM, 256, 0, stream>>>(x, wt, y);
}